// EquivariantMessagePassing_11020886081983
// MI455X (gfx1250) — compile-verified
//
#include <hip/hip_runtime.h>
#include <math.h>

#define MULC     64
#define NODE_DIM 576
#define NPATH    11
#define NTPW     704
#define TILE     16

typedef float v2f __attribute__((ext_vector_type(2)));
typedef float v4f __attribute__((ext_vector_type(4)));
typedef float v8f __attribute__((ext_vector_type(8)));
typedef int v4i_vs __attribute__((vector_size(4 * sizeof(int))));

__constant__ int c_pi[NPATH] = {0,0,0,1,1,1,1,2,2,2,2};
__constant__ int c_pj[NPATH] = {0,1,2,0,1,1,2,0,1,2,2};
__constant__ int c_pk[NPATH] = {0,1,2,1,0,2,1,2,1,0,2};

// ---------------------------------------------------------------------------
// fp32 WMMA wrapper: D(16x16) = A(16x4) * B(4x16) + C
// A/B layout per CDNA5 ISA 7.12.2: lane<16 holds K={0,1}, lane>=16 K={2,3}.
// ---------------------------------------------------------------------------
__device__ __forceinline__ v8f wmma_f32(v2f a, v2f b, v8f c) {
  return __builtin_amdgcn_wmma_f32_16x16x4_f32(false, a, false, b, (short)0, c,
                                               false, false);
}

// silu with hardware reciprocal (v_rcp_f32) instead of precise IEEE division
__device__ __forceinline__ float fast_silu(float x) {
  return x * __builtin_amdgcn_rcpf(1.0f + __expf(-x));
}

// ---------------------------------------------------------------------------
// Async global->LDS staging (CDNA5 GLOBAL_LOAD_ASYNC_TO_LDS_B128, ASYNCcnt).
// Builtin takes int4* operands per clang diagnostic. Guarded with fallback.
// ---------------------------------------------------------------------------
#if __has_builtin(__builtin_amdgcn_global_load_async_to_lds_b128)
#define HAVE_ASYNC_LDS 1
__device__ __forceinline__ void async_copy16(const float* g, float* l) {
  __builtin_amdgcn_global_load_async_to_lds_b128((v4i_vs*)g, (v4i_vs*)l, 0, 0);
}
__device__ __forceinline__ void async_wait0() {
#if __has_builtin(__builtin_amdgcn_s_wait_asynccnt)
  __builtin_amdgcn_s_wait_asynccnt(0);
#else
  asm volatile("s_wait_asynccnt 0x0" ::: "memory");
#endif
}
#else
#define HAVE_ASYNC_LDS 0
__device__ __forceinline__ void async_copy16(const float* g, float* l) {
  *(v4f*)l = *(const v4f*)g;
}
__device__ __forceinline__ void async_wait0() {}
#endif

// ===========================================================================
// Kernel 0: zero the output buffer (atomic aggregation target)
// ===========================================================================
__global__ void zero_kernel(float* __restrict__ p, long n) {
  long i = (long)blockIdx.x * blockDim.x + threadIdx.x;
  long stride = (long)gridDim.x * blockDim.x;
  for (; i < n; i += stride) p[i] = 0.0f;
}

// ===========================================================================
// Kernel 1: compute the 11 real Wigner-3j tensors on device (double precision)
// ws layout: float[NPATH][128], dense (a*db+b)*dc + c
// ===========================================================================
__device__ double dfact(int n) {
  double r = 1.0;
  for (int i = 2; i <= n; ++i) r *= (double)i;
  return r;
}

__device__ double su2cg(int j1, int j2, int j3, int m1, int m2, int m3) {
  if (m1 + m2 != m3) return 0.0;
  double pref = (2.0 * j3 + 1.0) * dfact(j1 + j2 - j3) * dfact(j1 - j2 + j3) *
                dfact(-j1 + j2 + j3) / dfact(j1 + j2 + j3 + 1) *
                dfact(j1 + m1) * dfact(j1 - m1) * dfact(j2 + m2) *
                dfact(j2 - m2) * dfact(j3 + m3) * dfact(j3 - m3);
  pref = sqrt(pref);
  double s = 0.0;
  for (int k = 0; k <= j1 + j2 + j3; ++k) {
    int d0 = k, d1 = j1 + j2 - j3 - k, d2 = j1 - m1 - k, d3 = j2 + m2 - k;
    int d4 = j3 - j2 + m1 + k, d5 = j3 - j1 - m2 + k;
    if (d0 < 0 || d1 < 0 || d2 < 0 || d3 < 0 || d4 < 0 || d5 < 0) continue;
    double den = dfact(d0) * dfact(d1) * dfact(d2) * dfact(d3) * dfact(d4) * dfact(d5);
    s += ((k & 1) ? -1.0 : 1.0) / den;
  }
  return pref * s;
}

__device__ void buildQ(int l, double qr[5][5], double qi[5][5]) {
  for (int a = 0; a < 5; a++)
    for (int b = 0; b < 5; b++) { qr[a][b] = 0.0; qi[a][b] = 0.0; }
  const double inv2 = 0.70710678118654752440;
  for (int m = -l; m < 0; ++m) { qr[l + m][l - m] = inv2; qi[l + m][l + m] = -inv2; }
  qr[l][l] = 1.0;
  for (int m = 1; m <= l; ++m) {
    double sg = (m & 1) ? -1.0 : 1.0;
    qr[l + m][l + m] = sg * inv2;
    qi[l + m][l - m] = sg * inv2;
  }
  double fr, fi;  // (-i)^l
  switch (l & 3) {
    case 0: fr = 1; fi = 0; break;
    case 1: fr = 0; fi = -1; break;
    case 2: fr = -1; fi = 0; break;
    default: fr = 0; fi = 1; break;
  }
  for (int a = 0; a < 2 * l + 1; ++a)
    for (int b = 0; b < 2 * l + 1; ++b) {
      double r = qr[a][b], im = qi[a][b];
      qr[a][b] = r * fr - im * fi;
      qi[a][b] = r * fi + im * fr;
    }
}

__global__ void w3j_setup_kernel(float* __restrict__ w3j) {
  int p = threadIdx.x;
  if (p >= NPATH) return;
  int l1 = c_pi[p], l2 = c_pj[p], l3 = c_pk[p];
  int d1 = 2 * l1 + 1, d2 = 2 * l2 + 1, d3 = 2 * l3 + 1;

  double cg[5][5][5];
  for (int i = 0; i < d1; i++)
    for (int k = 0; k < d2; k++)
      for (int n = 0; n < d3; n++)
        cg[i][k][n] = su2cg(l1, l2, l3, i - l1, k - l2, n - l3);

  double q1r[5][5], q1i[5][5], q2r[5][5], q2i[5][5], q3r[5][5], q3i[5][5];
  buildQ(l1, q1r, q1i);
  buildQ(l2, q2r, q2i);
  buildQ(l3, q3r, q3i);

  double Cr[5][5][5], Ci[5][5][5];
  double sre = 0.0, sim = 0.0;
  for (int a = 0; a < d1; a++)
    for (int b = 0; b < d2; b++)
      for (int c = 0; c < d3; c++) {
        double ar = 0.0, ai = 0.0;
        for (int i = 0; i < d1; i++)
          for (int k = 0; k < d2; k++) {
            double t1r = q1r[i][a] * q2r[k][b] - q1i[i][a] * q2i[k][b];
            double t1i = q1r[i][a] * q2i[k][b] + q1i[i][a] * q2r[k][b];
            for (int n = 0; n < d3; n++) {
              double g = cg[i][k][n];
              if (g == 0.0) continue;
              double qr3 = q3r[n][c], qi3 = -q3i[n][c];  // conj
              ar += g * (t1r * qr3 - t1i * qi3);
              ai += g * (t1r * qi3 + t1i * qr3);
            }
          }
        Cr[a][b][c] = ar;
        Ci[a][b][c] = ai;
        sre += fabs(ar);
        sim += fabs(ai);
      }
  bool useRe = (sre >= sim);
  double nrm = 0.0;
  for (int a = 0; a < d1; a++)
    for (int b = 0; b < d2; b++)
      for (int c = 0; c < d3; c++) {
        double v = useRe ? Cr[a][b][c] : Ci[a][b][c];
        nrm += v * v;
      }
  nrm = sqrt(nrm);
  if (nrm == 0.0) nrm = 1.0;
  float* dst = w3j + p * 128;
  for (int a = 0; a < d1; a++)
    for (int b = 0; b < d2; b++)
      for (int c = 0; c < d3; c++)
        dst[(a * d2 + b) * d3 + c] =
            (float)((useRe ? Cr[a][b][c] : Ci[a][b][c]) / nrm);
}

// ===========================================================================
// Kernel 2: per-edge MLP (fp32 WMMA) + tensor product + atomic scatter
// One wave32 per tile of 16 edges. Gathered node rows are staged to LDS with
// async global->LDS copies, hidden behind the WMMA MLP.
// ===========================================================================
__global__ void __launch_bounds__(32)
edge_kernel(const float* __restrict__ node_feat,
            const int* __restrict__ edge_index,
            const float* __restrict__ edge_vec,
            const float* __restrict__ fc_w0,
            const float* __restrict__ fc_w1,
            const float* __restrict__ fc_w2,
            const float* __restrict__ w3j,
            float* __restrict__ out, int E) {
  __shared__ float s_h0[TILE][64];
  __shared__ float s_h1[TILE][64];
  __shared__ float s_w[TILE][64];
  __shared__ float s_msg[TILE][NODE_DIM];
  __shared__ float s_x[TILE][NODE_DIM];   // gathered node_feat[src] rows
  __shared__ float s_sh[TILE][9];
  __shared__ float s_dist[TILE];
  __shared__ int s_src[TILE], s_dst[TILE];

  const int lane = threadIdx.x;
  const int half = lane >> 4;
  const int l15 = lane & 15;
  const int base = blockIdx.x * TILE;

  // --- per-edge geometry + spherical harmonics (lanes 0..15) ---
  if (lane < TILE) {
    int e = base + lane;
    if (e >= E) e = E - 1;
    float ex = edge_vec[3 * e + 0], ey = edge_vec[3 * e + 1], ez = edge_vec[3 * e + 2];
    float d = sqrtf(ex * ex + ey * ey + ez * ez);
    float inv = 1.0f / fmaxf(d, 1e-9f);
    float x = ex * inv, y = ey * inv, z = ez * inv;
    const float s3 = 1.7320508075688772f;
    const float s5 = 2.2360679774997896f;
    s_sh[lane][0] = 1.0f;
    s_sh[lane][1] = s3 * y;
    s_sh[lane][2] = s3 * z;
    s_sh[lane][3] = s3 * x;
    s_sh[lane][4] = s5 * (s3 * x * y);
    s_sh[lane][5] = s5 * (s3 * y * z);
    s_sh[lane][6] = s5 * (1.5f * z * z - 0.5f);
    s_sh[lane][7] = s5 * (s3 * x * z);
    s_sh[lane][8] = s5 * (0.5f * s3 * (x * x - y * y));
    s_dist[lane] = d;
    s_src[lane] = edge_index[e];
    s_dst[lane] = edge_index[E + e];
  }
  // zero message accumulator
  for (int idx = lane; idx < TILE * NODE_DIM; idx += 32)
    (&s_msg[0][0])[idx] = 0.0f;
  __syncthreads();

  // --- kick off async gather of the 16 src-node rows into LDS ------------
  // 576 floats per row = 144 x 16B copies; latency hidden by the MLP below.
  for (int idx = lane; idx < TILE * (NODE_DIM / 4); idx += 32) {
    int e = idx / (NODE_DIM / 4);
    int c4 = idx - e * (NODE_DIM / 4);
    async_copy16(node_feat + (size_t)s_src[e] * NODE_DIM + c4 * 4,
                 &s_x[e][c4 * 4]);
  }

  // --- h0 = silu(dist * w0) : 16x64 tile in LDS ---
  for (int idx = lane; idx < TILE * 64; idx += 32) {
    int r = idx >> 6, c = idx & 63;
    (&s_h0[0][0])[idx] = fast_silu(s_dist[r] * fc_w0[c]);
  }
  __syncthreads();

  // --- GEMM1: h1 = silu((h0 @ W1) * 0.125), fp32 WMMA 16x16x4 ---
  for (int t = 0; t < 4; ++t) {
    v8f acc;
    for (int i = 0; i < 8; ++i) acc[i] = 0.0f;
    for (int s = 0; s < 16; ++s) {
      int k0 = 4 * s + 2 * half;
      v2f a;
      a.x = s_h0[l15][k0];
      a.y = s_h0[l15][k0 + 1];
      v2f b;
      b.x = fc_w1[k0 * 64 + t * 16 + l15];
      b.y = fc_w1[(k0 + 1) * 64 + t * 16 + l15];
      acc = wmma_f32(a, b, acc);
    }
    for (int v = 0; v < 8; ++v) {
      int row = v + 8 * half;  // D layout: VGPR v -> M=v (lanes<16) / v+8
      s_h1[row][t * 16 + l15] = fast_silu(acc[v] * 0.125f);
    }
  }
  // gathered rows must be resident before the TP loop reads them
  async_wait0();
  __syncthreads();

  // --- per-path: 64-col tp_w chunk (WMMA) + CG contraction ---
  for (int p = 0; p < NPATH; ++p) {
    const int li = c_pi[p], lj = c_pj[p], lk = c_pk[p];
    const int da = 2 * li + 1, db = 2 * lj + 1, dc = 2 * lk + 1;

    for (int t = 0; t < 4; ++t) {
      v8f acc;
      for (int i = 0; i < 8; ++i) acc[i] = 0.0f;
      for (int s = 0; s < 16; ++s) {
        int k0 = 4 * s + 2 * half;
        v2f a;
        a.x = s_h1[l15][k0];
        a.y = s_h1[l15][k0 + 1];
        int col = p * 64 + t * 16 + l15;
        v2f b;
        b.x = fc_w2[k0 * NTPW + col];
        b.y = fc_w2[(k0 + 1) * NTPW + col];
        acc = wmma_f32(a, b, acc);
      }
      for (int v = 0; v < 8; ++v)
        s_w[v + 8 * half][t * 16 + l15] = acc[v] * 0.125f;
    }
    __syncthreads();

    const float* C = w3j + p * 128;
    const int offi = (li == 0) ? 0 : ((li == 1) ? 64 : 256);
    const int offk = (lk == 0) ? 0 : ((lk == 1) ? 64 : 256);
    const int offsj = (lj == 0) ? 0 : ((lj == 1) ? 1 : 4);
    const int np = (lk == 0) ? 3 : 4;
    const float alpha = sqrtf((float)(2 * lk + 1) / (float)np);

    for (int pr = lane; pr < TILE * 64; pr += 32) {
      int e = pr >> 6, u = pr & 63;
      const float* xr = &s_x[e][offi + u * da];
      float wv = s_w[e][u] * alpha;
      float oc[5] = {0.f, 0.f, 0.f, 0.f, 0.f};
      for (int a = 0; a < da; ++a) {
        float xa = xr[a];
        for (int b = 0; b < db; ++b) {
          float xs = xa * s_sh[e][offsj + b];
          const float* Crow = C + (a * db + b) * dc;
          for (int c = 0; c < dc; ++c) oc[c] += Crow[c] * xs;
        }
      }
      float* m = &s_msg[e][offk + u * dc];
      for (int c = 0; c < dc; ++c) m[c] += wv * oc[c];
    }
    __syncthreads();
  }

  // --- scatter-add messages to destination nodes (f32 atomics in L2) ---
  for (int idx = lane; idx < TILE * NODE_DIM; idx += 32) {
    int e = idx / NODE_DIM, f = idx - e * NODE_DIM;
    if (base + e < E)
      unsafeAtomicAdd(&out[(size_t)s_dst[e] * NODE_DIM + f],
                      (&s_msg[0][0])[idx]);
  }
}

// ===========================================================================
// Kernel 3: per-node channel mix (lin_w) + residual, in place on d_out
// ===========================================================================
__global__ void __launch_bounds__(64)
node_kernel(const float* __restrict__ node_feat, const float* __restrict__ lin_w,
            float* __restrict__ out, int N) {
  int n = blockIdx.x;
  int v = threadIdx.x;  // output channel
  __shared__ float a[NODE_DIM];
  for (int f = v; f < NODE_DIM; f += 64) a[f] = out[(size_t)n * NODE_DIM + f];
  __syncthreads();
  float res[9];
  int idx = 0;
  for (int l = 0; l < 3; ++l) {
    int dl = 2 * l + 1;
    int off = (l == 0) ? 0 : ((l == 1) ? 64 : 256);
    const float* W = lin_w + l * 4096;
    for (int d = 0; d < dl; ++d) {
      float s = 0.0f;
      for (int u = 0; u < 64; ++u) s += a[off + u * dl + d] * W[u * 64 + v];
      res[idx++] = s * 0.125f;
    }
  }
  idx = 0;
  for (int l = 0; l < 3; ++l) {
    int dl = 2 * l + 1;
    int off = (l == 0) ? 0 : ((l == 1) ? 64 : 256);
    for (int d = 0; d < dl; ++d) {
      size_t o = (size_t)n * NODE_DIM + off + v * dl + d;
      out[o] = res[idx++] + node_feat[o];
    }
  }
}

// ===========================================================================
extern "C" void kernel_launch(void* const* d_in, const int* in_sizes, int n_in,
                              void* d_out, int out_size, void* d_ws,
                              size_t ws_size, hipStream_t stream) {
  const float* node_feat = (const float*)d_in[0];
  const int* edge_index = (const int*)d_in[1];
  const float* edge_vec = (const float*)d_in[2];
  const float* fc_w0 = (const float*)d_in[3];
  const float* fc_w1 = (const float*)d_in[4];
  const float* fc_w2 = (const float*)d_in[5];
  const float* lin_w = (const float*)d_in[6];
  float* out = (float*)d_out;

  const int N = in_sizes[0] / NODE_DIM;
  const int E = in_sizes[2] / 3;
  float* w3j = (float*)d_ws;  // NPATH*128 floats

  long total = (long)N * NODE_DIM;
  zero_kernel<<<1024, 256, 0, stream>>>(out, total);
  w3j_setup_kernel<<<1, 32, 0, stream>>>(w3j);
  edge_kernel<<<(E + TILE - 1) / TILE, 32, 0, stream>>>(
      node_feat, edge_index, edge_vec, fc_w0, fc_w1, fc_w2, w3j, out, E);
  node_kernel<<<N, 64, 0, stream>>>(node_feat, lin_w, out, N);
}